// QAEModel_71116068487241
// MI455X (gfx1250) — compile-verified
//
#include <hip/hip_runtime.h>
#include <hip/hip_bf16.h>

// ---------------------------------------------------------------------------
// QAE circuit: 8 variational qubits (ref qubits never touched -> 256-dim sim).
//   Step 0: zero output + precompute cos/sin(theta/2) tables (168 each).
//   Step 1: build the 256x256 circuit operator U (theta-dependent only).
//   Step 2: Y = X(1024x256) @ U(256x256) with V_WMMA_F32_16X16X4_F32 (fp32).
//   Step 3: out[b][j] = sum_a Y[b, 8a+j]^2 via shuffle pre-reduce + f32 atomics.
// ---------------------------------------------------------------------------

typedef __attribute__((ext_vector_type(2))) float v2f;
typedef __attribute__((ext_vector_type(8))) float v8f;

#define NQV     8      // variational qubits
#define DIM8    256    // 2^NQV
#define MLAYERS 20
#define NANG    ((MLAYERS + 1) * NQV)   // 168 gate angles
#define BATCH   1024

// Workspace layout (floats): [0, 256*256) = U ; then ctab[168], stab[168].
#define WS_U    0
#define WS_CTAB (DIM8 * DIM8)
#define WS_STAB (WS_CTAB + NANG)

// ---------------------------------------------------------------------------
// Kernel 0: zero the atomic-accumulated output and build the trig tables once
// (instead of 65536 threads each redundantly evaluating 168 cos/sin pairs).
// ---------------------------------------------------------------------------
__global__ void init_kernel(const float* __restrict__ theta,
                            float* __restrict__ ws,
                            float* __restrict__ out, int out_n) {
  const int i = blockIdx.x * blockDim.x + threadIdx.x;
  if (i < out_n) out[i] = 0.0f;
  if (i < NANG) {
    const float half = 0.5f * theta[i];
    ws[WS_CTAB + i] = cosf(half);
    ws[WS_STAB + i] = sinf(half);
  }
}

// ---------------------------------------------------------------------------
// Kernel 1: build U. Block r simulates basis state e_r through the circuit.
// Thread tid owns amplitude tid of the 256-dim state (qubit q <-> bit 7-q).
// RY with pair-stride < 32: wave32 shfl_xor (no barrier).
// RY with pair-stride >= 32 and the fused CNOT-ring permutation: LDS.
// Gate cos/sin come from the uniform tables -> scalar loads, no per-lane trig.
// ---------------------------------------------------------------------------
__global__ __launch_bounds__(256) void build_U(const float* __restrict__ ws,
                                               float* __restrict__ U) {
  __shared__ float lds[DIM8];
  const float* __restrict__ ctab = ws + WS_CTAB;
  const float* __restrict__ stab = ws + WS_STAB;
  const int tid = threadIdx.x;
  const int r   = blockIdx.x;

  float a = (tid == r) ? 1.0f : 0.0f;

  // Fuse the whole CNOT ring (q -> (q+1)%8, q = 0..7 in order) into one
  // gather: state'[i] = state[g0(g1(...g7(i)))], gq(i) = i ^ (bit_pc(i)<<pt).
  // Same permutation every layer -> precompute per thread.
  int src = tid;
#pragma unroll
  for (int q = NQV - 1; q >= 0; --q) {
    const int pc = (NQV - 1) - q;                     // control bit position
    const int pt = (NQV - 1) - ((q + 1) & (NQV - 1)); // target bit position
    src ^= ((src >> pc) & 1) << pt;
  }

  for (int m = 0; m <= MLAYERS; ++m) {
    // ---- RY layer: out0 = c*a0 + s*a1 ; out1 = -s*a0 + c*a1 ----
#pragma unroll
    for (int q = 0; q < NQV; ++q) {
      const float c = ctab[m * NQV + q];   // wave-uniform
      const float s = stab[m * NQV + q];
      const int mask = 1 << ((NQV - 1) - q);
      float p;
      if (mask >= 32) {                 // partner in another wave -> LDS
        __syncthreads();
        lds[tid] = a;
        __syncthreads();
        p = lds[tid ^ mask];
      } else {                          // partner inside wave32 -> shuffle
        p = __shfl_xor(a, mask, 32);
      }
      a = (tid & mask) ? (c * a - s * p) : (c * a + s * p);
    }
    // ---- fused CNOT ring (skipped after final RY layer) ----
    if (m < MLAYERS) {
      __syncthreads();
      lds[tid] = a;
      __syncthreads();
      a = lds[src];
    }
  }

  U[r * DIM8 + tid] = a;   // row r of U: Y = X @ U
}

// ---------------------------------------------------------------------------
// Kernel 2: one wave32 per 16x16 output tile of Y = X @ U, K = 256 in steps
// of 4 via V_WMMA_F32_16X16X4_F32. Epilogue squares the fragment, folds the
// two a-groups per tile with shfl_xor(8), and atomically accumulates into
// out[b][j] (j = column & 7).
// ---------------------------------------------------------------------------
__global__ __launch_bounds__(256) void gemm_probs(const float* __restrict__ X,
                                                  const float* __restrict__ U,
                                                  float* __restrict__ out) {
  const int lane  = threadIdx.x & 31;
  const int wave  = threadIdx.x >> 5;
  const int tile  = blockIdx.x * 8 + wave;  // 1024 tiles = 64 Mtiles x 16 Ntiles
  const int m0    = (tile >> 4) * 16;
  const int n0    = (tile & 15) * 16;

  const int l15   = lane & 15;
  const int khalf = (lane >> 4) << 1;       // lanes 0-15 -> K{0,1}; 16-31 -> K{2,3}

  const float* __restrict__ xrow = X + (m0 + l15) * DIM8;   // A: M = lane&15
  const float* __restrict__ ucol = U + (n0 + l15);          // B: N = lane&15

  v8f acc = {};
  for (int k = 0; k < DIM8; k += 4) {
    const int ka = k + khalf;
    v2f afrag;                     // A 16x4 f32: v0 = K even-slot, v1 = next K
    afrag.x = xrow[ka];
    afrag.y = xrow[ka + 1];
    v2f bfrag;                     // B 4x16 f32: same K striping across lanes
    bfrag.x = ucol[ka * DIM8];
    bfrag.y = ucol[(ka + 1) * DIM8];
    acc = __builtin_amdgcn_wmma_f32_16x16x4_f32(
        /*neg_a=*/false, afrag, /*neg_b=*/false, bfrag,
        /*c_mod=*/(short)0, acc, /*reuse_a=*/false, /*reuse_b=*/false);
  }

  // C/D layout: VGPR v, lanes 0-15 -> (M=v, N=lane); lanes 16-31 -> (M=v+8).
  // Columns N and N^8 share j = N&7 (different a-group) -> fold with shuffle.
  float partial[8];
#pragma unroll
  for (int v = 0; v < 8; ++v) {
    const float y  = acc[v];
    float sq = y * y;
    sq += __shfl_xor(sq, 8, 32);
    partial[v] = sq;
  }
  if (((lane >> 3) & 1) == 0) {
    const int j = lane & 7;
#pragma unroll
    for (int v = 0; v < 8; ++v) {
      const int mrow = m0 + v + ((lane >> 4) << 3);
      atomicAdd(&out[mrow * 8 + j], partial[v]);
    }
  }
}

extern "C" void kernel_launch(void* const* d_in, const int* in_sizes, int n_in,
                              void* d_out, int out_size, void* d_ws, size_t ws_size,
                              hipStream_t stream) {
  const float* x     = (const float*)d_in[0];   // [1024, 256] f32
  const float* theta = (const float*)d_in[1];   // [168] f32
  float* out = (float*)d_out;                   // [1024, 8] f32
  float* ws  = (float*)d_ws;                    // U (256KB) + trig tables

  init_kernel<<<(out_size + 255) / 256, 256, 0, stream>>>(theta, ws, out, out_size);
  build_U<<<DIM8, 256, 0, stream>>>(ws, ws + WS_U);
  gemm_probs<<<128, 256, 0, stream>>>(x, ws + WS_U, out);   // 1024 tiles / 8 waves
}